// ROUGELoss_49443663511731
// MI455X (gfx1250) — compile-verified
//
#include <hip/hip_runtime.h>
#include <hip/hip_bf16.h>

typedef float v2f __attribute__((ext_vector_type(2)));
typedef float v4f __attribute__((ext_vector_type(4)));
typedef float v8f __attribute__((ext_vector_type(8)));

#define BB 16
#define LL 128
#define LR 128
#define VV 32000

// ---------------------------------------------------------------------------
// Kernel 1: one block per (b,j) row of logits. Single streaming pass:
// online softmax (running max m, running sum s of exp(x-m)) + argmax.
// Non-temporal b128 loads: 262MB of single-use data, don't pollute caches.
// ---------------------------------------------------------------------------
__global__ __launch_bounds__(256) void rouge_row_stats(
    const float* __restrict__ logits,
    float* __restrict__ mx, float* __restrict__ sm, int* __restrict__ am)
{
    __shared__ float red_m[256];
    __shared__ float red_s[256];
    __shared__ int   red_a[256];

    const int r   = blockIdx.x;            // 0 .. B*L-1
    const int tid = threadIdx.x;
    const v4f* row = (const v4f*)(logits + (size_t)r * VV);

    float m = -__builtin_inff();
    float s = 0.0f;
    int   a = 0;

    for (int idx = tid; idx < VV / 4; idx += 256) {
        v4f x4 = __builtin_nontemporal_load(&row[idx]);
        float xs[4] = {x4.x, x4.y, x4.z, x4.w};
        #pragma unroll
        for (int c = 0; c < 4; ++c) {
            float x = xs[c];
            if (x > m) {                       // rare after warm-up: ~1 exp/elem
                s = s * __expf(m - x) + 1.0f;  // exp(-inf)==0 handles init
                m = x;
                a = idx * 4 + c;
            } else {
                s += __expf(x - m);
            }
        }
    }

    red_m[tid] = m; red_s[tid] = s; red_a[tid] = a;
    __syncthreads();

    for (int st = 128; st > 0; st >>= 1) {
        if (tid < st) {
            float m1 = red_m[tid],      s1 = red_s[tid];      int a1 = red_a[tid];
            float m2 = red_m[tid + st], s2 = red_s[tid + st]; int a2 = red_a[tid + st];
            float mo, so; int ao;
            if (m2 > m1)      { mo = m2; so = s2 + s1 * __expf(m1 - m2); ao = a2; }
            else if (m1 > m2) { mo = m1; so = s1 + s2 * __expf(m2 - m1); ao = a1; }
            else              { mo = m1; so = s1 + s2; ao = (a1 < a2) ? a1 : a2; } // first-index tie
            red_m[tid] = mo; red_s[tid] = so; red_a[tid] = ao;
        }
        __syncthreads();
    }

    if (tid == 0) { mx[r] = red_m[0]; sm[r] = red_s[0]; am[r] = red_a[0]; }
}

// ---------------------------------------------------------------------------
// Kernel 2: one block (256 thr = 8 waves) per batch.
//  - 128-bit match masks per label row; keep lowest set bit (row dedup)
//  - sequential column dedup (128 tiny steps, thread 0)
//  - weighted sum of probs via V_WMMA_F32_16X16X4_F32 with all-ones B:
//    fully branchless inner loop (EXEC stays all-1s), masks hoisted to regs.
// ---------------------------------------------------------------------------
__global__ __launch_bounds__(256) void rouge_batch_sum(
    const float* __restrict__ logits, const int* __restrict__ labels,
    const float* __restrict__ mx, const float* __restrict__ sm,
    const int* __restrict__ am, float* __restrict__ Sb)
{
    __shared__ int      lab_s[LR];
    __shared__ int      hard_s[LL];
    __shared__ float    stat_s[LL][2];   // [j][0]=rowmax, [j][1]=1/rowsum
    __shared__ unsigned rf_s[LR][4];     // row-first masks
    __shared__ unsigned fin_s[LR][4];    // final dedup masks
    __shared__ unsigned rowflag_s[LR];
    __shared__ unsigned colmask_s[4];
    __shared__ float    wsum_s[8];

    const int b   = blockIdx.x;
    const int tid = threadIdx.x;

    if (tid < LL) {
        const int r = b * LL + tid;
        lab_s[tid]     = labels[r];
        hard_s[tid]    = am[r];
        stat_s[tid][0] = mx[r];
        stat_s[tid][1] = 1.0f / sm[r];
    }
    __syncthreads();

    if (tid < LR) {
        const int li = lab_s[tid];
        #pragma unroll
        for (int w = 0; w < 4; ++w) {
            unsigned bits = 0u;
            for (int k = 0; k < 32; ++k)
                bits |= ((unsigned)(li == hard_s[w * 32 + k])) << k;
            rf_s[tid][w] = bits;
        }
        // isolate lowest set bit across the 128-bit mask (first match in row)
        bool found = false;
        #pragma unroll
        for (int w = 0; w < 4; ++w) {
            unsigned x = rf_s[tid][w];
            rf_s[tid][w] = found ? 0u : (x & (0u - x));
            found = found || (x != 0u);
        }
    }
    __syncthreads();

    if (tid == 0) {  // column dedup: keep first row claiming each column
        unsigned used[4] = {0u, 0u, 0u, 0u};
        for (int i = 0; i < LR; ++i) {
            unsigned any = 0u;
            #pragma unroll
            for (int w = 0; w < 4; ++w) {
                unsigned f = rf_s[i][w] & ~used[w];
                fin_s[i][w] = f;
                any |= f;
                used[w] |= rf_s[i][w];
            }
            rowflag_s[i] = (any != 0u) ? 1u : 0u;
        }
        #pragma unroll
        for (int w = 0; w < 4; ++w) colmask_s[w] = used[w];  // OR(final) == OR(row_first)
    }
    __syncthreads();

    // --- WMMA weighted reduction: wave handles a 16-row block, K=4 col chunks.
    // A-matrix f32 16x4 layout: lanes 0-15 -> M=lane, K=0,1 ; lanes 16-31 -> M=lane-16, K=2,3
    const int wave = tid >> 5;
    const int lane = tid & 31;
    const int i    = wave * 16 + (lane & 15);
    const int joff = (lane >> 4) * 2;
    const int li   = lab_s[i];
    const float rnot = rowflag_s[i] ? 0.0f : 1.0f;   // (1 - rowflag)
    const size_t lbase = (size_t)b * LL * VV + (size_t)li;

    // hoist this lane's mask words into registers (no LDS in inner loop)
    unsigned finw[4], colw[4];
    #pragma unroll
    for (int w = 0; w < 4; ++w) { finw[w] = fin_s[i][w]; colw[w] = colmask_s[w]; }

    v8f acc   = {0.f, 0.f, 0.f, 0.f, 0.f, 0.f, 0.f, 0.f};
    v2f onesB = {1.0f, 1.0f};

    #pragma unroll
    for (int c = 0; c < LL; c += 4) {
        // word index is compile-time constant: (c+joff+t)>>5 == c>>5 (c%32 <= 28, joff+t <= 3)
        const unsigned fw = finw[c >> 5];
        const unsigned cw = colw[c >> 5];
        v2f aval;
        #pragma unroll
        for (int t = 0; t < 2; ++t) {
            const int j = c + joff + t;
            const float g = logits[lbase + (size_t)j * VV];       // logits[b, j, lab[i]]
            const float p = __expf(g - stat_s[j][0]) * stat_s[j][1];
            const float bitf = (float)((fw >> (j & 31)) & 1u);
            const float colf = (float)((cw >> (j & 31)) & 1u);
            // matched -> 1.0 ; both-unmatched -> 0.5 ; else 0.1   (branchless)
            const float w = 0.1f + 0.9f * bitf + 0.4f * rnot * (1.0f - colf);
            aval[t] = w * p;
        }
        // D[m,n] += sum_k A[m,k] * 1  (row sums replicated over 16 N columns)
        acc = __builtin_amdgcn_wmma_f32_16x16x4_f32(
            false, aval, false, onesB, (short)0, acc, false, false);
    }

    float local = acc[0] + acc[1] + acc[2] + acc[3]
                + acc[4] + acc[5] + acc[6] + acc[7];
    for (int o = 16; o > 0; o >>= 1)
        local += __shfl_down(local, o, 32);
    if (lane == 0) wsum_s[wave] = local * (1.0f / 16.0f);  // each rowsum counted 16x
    __syncthreads();

    if (tid == 0) {
        float s = 0.0f;
        #pragma unroll
        for (int w = 0; w < 8; ++w) s += wsum_s[w];
        Sb[b] = s;
    }
}

// ---------------------------------------------------------------------------
// Kernel 3: finalize  out = 1 + mean_b(-2*S_b / 256) = 1 - sum(S_b)/2048
// ---------------------------------------------------------------------------
__global__ void rouge_final(const float* __restrict__ Sb, float* __restrict__ out)
{
    if (threadIdx.x == 0 && blockIdx.x == 0) {
        float s = 0.0f;
        #pragma unroll
        for (int b = 0; b < BB; ++b) s += Sb[b];
        out[0] = 1.0f - s * (1.0f / 2048.0f);
    }
}

extern "C" void kernel_launch(void* const* d_in, const int* in_sizes, int n_in,
                              void* d_out, int out_size, void* d_ws, size_t ws_size,
                              hipStream_t stream)
{
    const float* logits = (const float*)d_in[0];   // [16,128,32000] f32
    const int*   labels = (const int*)d_in[1];     // [16,128] int

    char* ws = (char*)d_ws;
    float* mx = (float*)(ws);                      // 2048 f32
    float* sm = (float*)(ws + 2048 * 4);           // 2048 f32
    int*   am = (int*)  (ws + 2048 * 8);           // 2048 i32
    float* Sb = (float*)(ws + 2048 * 12);          // 16 f32

    rouge_row_stats<<<BB * LL, 256, 0, stream>>>(logits, mx, sm, am);
    rouge_batch_sum<<<BB, 256, 0, stream>>>(logits, labels, mx, sm, am, Sb);
    rouge_final<<<1, 32, 0, stream>>>(Sb, (float*)d_out);
}